// UDecoder_31894427140827
// MI455X (gfx1250) — compile-verified
//
#include <hip/hip_runtime.h>
#include <math.h>

typedef float v2f __attribute__((ext_vector_type(2)));
typedef float v8f __attribute__((ext_vector_type(8)));
typedef int   i32x4_vs __attribute__((vector_size(16)));   // matches builtin param type

#define C_Kc   64
#define C_Hc   256
#define C_OUTc 80
#define NHEADc 4
#define ALPHAc 5.0f
#define LVLc   5
#define Bc     16
#define Sc     512

#define MODE_ADD  1
#define MODE_GELU 2

#define AS1 __attribute__((address_space(1)))
#define AS3 __attribute__((address_space(3)))

// CDNA5 async global->LDS path (ASYNCcnt-tracked), guarded so the file still
// compiles (with the VGPR-bounce fallback) if the builtins are not declared.
#if defined(__has_builtin)
#if __has_builtin(__builtin_amdgcn_global_load_async_to_lds_b32) && \
    __has_builtin(__builtin_amdgcn_global_load_async_to_lds_b128) && \
    __has_builtin(__builtin_amdgcn_s_wait_asynccnt)
#define HAVE_ASYNC_LDS 1
#endif
#endif
#ifndef HAVE_ASYNC_LDS
#define HAVE_ASYNC_LDS 0
#endif

__device__ __forceinline__ float gelu_f(float x) {
    // exact GELU: 0.5*x*(1+erf(x/sqrt(2)))
    return 0.5f * x * (1.0f + erff(x * 0.70710678118654752f));
}

// ---------------- L2 normalize per 16-channel head ----------------
// src[b][c][p] (c in 0..63 of a possibly larger channel block), dst[b][64][p]
__global__ void l2norm_heads_kernel(const float* __restrict__ src, float* __restrict__ dst,
                                    int P, long srcBatchStride) {
    long gid = (long)blockIdx.x * blockDim.x + threadIdx.x;
    if (gid >= (long)Bc * P) return;
    int b = (int)(gid / P), p = (int)(gid % P);
    const float* s = src + (long)b * srcBatchStride;
    float* d = dst + (long)b * C_Kc * P;
    for (int h = 0; h < NHEADc; ++h) {
        float tmp[16]; float ss = 0.f;
        #pragma unroll
        for (int c = 0; c < 16; ++c) {
            float v = s[(long)(h * 16 + c) * P + p];
            tmp[c] = v; ss += v * v;
        }
        float inv = 1.0f / fmaxf(sqrtf(ss), 1e-12f);
        #pragma unroll
        for (int c = 0; c < 16; ++c)
            d[(long)(h * 16 + c) * P + p] = tmp[c] * inv;
    }
}

// ---------------- attention: one wave per (b, h, t), flash-style ----------------
// qn[b][64][T], kn[b][64][512], kv raw [b][64+cwz][512] (v = rows 64..), z[b][cwz][T]
__global__ void attn_kernel(const float* __restrict__ qn, const float* __restrict__ kn,
                            const float* __restrict__ kv, float* __restrict__ z,
                            int T, int cwz) {
    int tid = threadIdx.x;
    int lane = tid & 31;
    int gw = blockIdx.x * (blockDim.x >> 5) + (tid >> 5);
    int total = Bc * NHEADc * T;
    if (gw >= total) return;
    int b = gw / (NHEADc * T);
    int rem = gw % (NHEADc * T);
    int h = rem / T;
    int t = rem % T;

    float qv = (lane < 16) ? qn[((long)b * C_Kc + h * 16 + lane) * T + t] : 0.f;
    const float* knb = kn + ((long)b * C_Kc + h * 16) * Sc;

    float sc[16];
    #pragma unroll
    for (int ch = 0; ch < 16; ++ch) {
        int s = ch * 32 + lane;
        float dot = 0.f;
        #pragma unroll
        for (int c = 0; c < 16; ++c) {
            float qc = __shfl(qv, c, 32);
            dot += qc * knb[(long)c * Sc + s];
        }
        sc[ch] = dot * ALPHAc;
    }
    float m = sc[0];
    #pragma unroll
    for (int i = 1; i < 16; ++i) m = fmaxf(m, sc[i]);
    for (int off = 16; off >= 1; off >>= 1) m = fmaxf(m, __shfl_xor(m, off, 32));
    float sum = 0.f;
    #pragma unroll
    for (int i = 0; i < 16; ++i) { sc[i] = __expf(sc[i] - m); sum += sc[i]; }
    for (int off = 16; off >= 1; off >>= 1) sum += __shfl_xor(sum, off, 32);
    float inv = 1.0f / sum;

    int cvh = cwz / NHEADc;
    const float* vb = kv + (long)b * (C_Kc + cwz) * Sc + (long)(C_Kc + h * cvh) * Sc;
    for (int cv = 0; cv < cvh; ++cv) {
        float acc = 0.f;
        #pragma unroll
        for (int ch = 0; ch < 16; ++ch)
            acc += sc[ch] * vb[(long)cv * Sc + ch * 32 + lane];
        for (int off = 16; off >= 1; off >>= 1) acc += __shfl_xor(acc, off, 32);
        if (lane == 0)
            z[((long)b * cwz + h * cvh + cv) * T + t] = acc * inv;
    }
}

// ---------------- 1x1 conv (GEMM) via f32 WMMA 16x16x4 ----------------
// Y[b][OC][T] (op)= W[OC][IC] @ X[b][IC][T] + bias, per-mode epilogue.
// Block = 256 threads = 8 waves; block covers 16 tokens x 128 out-channels.
__global__ void proj_wmma_kernel(const float* __restrict__ X, long xBatchStride,
                                 const float* __restrict__ W, const float* __restrict__ bias,
                                 float* __restrict__ Y, long yBatchStride,
                                 int IC, int T, int OC, int mode) {
    __shared__ float ldsX[C_Hc * 16];
    int tid = threadIdx.x;
    int tilesPerBatch = T >> 4;
    int b = blockIdx.x / tilesPerBatch;
    int t0 = (blockIdx.x % tilesPerBatch) << 4;

    const float* xb = X + (long)b * xBatchStride;
#if HAVE_ASYNC_LDS
    // Direct global->LDS async copy, 16B per lane (all addresses 16B-aligned:
    // T and t0 are multiples of 16, buffer bases are 16B-aligned).
    for (int idx = tid; idx < IC * 4; idx += 256) {
        int ci = idx >> 2, j = (idx & 3) << 2;
        __builtin_amdgcn_global_load_async_to_lds_b128(
            (AS1 i32x4_vs*)(xb + (long)ci * T + t0 + j),
            (AS3 i32x4_vs*)&ldsX[ci * 16 + j], 0, 0);
    }
    __builtin_amdgcn_s_wait_asynccnt(0);
#else
    for (int idx = tid; idx < IC * 16; idx += 256) {
        int ci = idx >> 4, j = idx & 15;
        ldsX[idx] = xb[(long)ci * T + t0 + j];
    }
#endif
    __syncthreads();

    int lane = tid & 31, wv = tid >> 5;
    int oc0 = blockIdx.y * 128 + wv * 16;
    if (oc0 >= OC) return;
    int r = lane & 15, hi = lane >> 4;

    v8f acc = {0.f, 0.f, 0.f, 0.f, 0.f, 0.f, 0.f, 0.f};
    // A fragment (W 16x4): lanes 0-15 -> K=k0,k0+1 ; lanes 16-31 -> K=k0+2,k0+3
    const float* Wrow = W + (long)(oc0 + r) * IC + hi * 2;
    __builtin_prefetch(Wrow, 0, 3);
    for (int k0 = 0; k0 < IC; k0 += 4) {
        v2f a, bb;
        a.x = Wrow[k0];
        a.y = Wrow[k0 + 1];
        int kk = k0 + hi * 2;
        bb.x = ldsX[kk * 16 + r];
        bb.y = ldsX[(kk + 1) * 16 + r];
        acc = __builtin_amdgcn_wmma_f32_16x16x4_f32(false, a, false, bb, (short)0, acc,
                                                    false, false);
    }

    float* yb = Y + (long)b * yBatchStride;
    #pragma unroll
    for (int g = 0; g < 8; ++g) {
        int oc = oc0 + g + 8 * hi;            // D layout: VGPR g, lane>=16 -> M=g+8
        float v = acc[g] + bias[oc];
        if (mode & MODE_GELU) v = gelu_f(v);
        long off = (long)oc * T + t0 + r;
        if (mode & MODE_ADD) yb[off] += v;
        else                 yb[off] = v;
    }
}

// ---------------- 3-tap conv with reflect pad, fused gelu + add-to-Y ----------------
// Y[b][256][T] += gelu( conv3(rs) + bias )   (Y preloaded with the cond branch)
__global__ void conv3_wmma_kernel(const float* __restrict__ Xrs,
                                  const float* __restrict__ W3, const float* __restrict__ bias,
                                  float* __restrict__ Y, int T) {
    __shared__ float ldsX[C_Hc * 18];
    int tid = threadIdx.x;
    int tilesPerBatch = T >> 4;
    int b = blockIdx.x / tilesPerBatch;
    int t0 = (blockIdx.x % tilesPerBatch) << 4;

    const float* xb = Xrs + (long)b * C_Hc * T;
#if HAVE_ASYNC_LDS
    // Reflect-indexed halo columns are only 4B-aligned -> use b32 async copies.
    for (int idx = tid; idx < C_Hc * 18; idx += 256) {
        int ci = idx / 18, j = idx % 18;
        int tg = t0 + j - 1;
        if (tg < 0) tg = -tg;                 // reflect: -1 -> 1
        if (tg >= T) tg = 2 * T - 2 - tg;     // reflect: T -> T-2
        __builtin_amdgcn_global_load_async_to_lds_b32(
            (AS1 int*)(xb + (long)ci * T + tg),
            (AS3 int*)&ldsX[idx], 0, 0);
    }
    __builtin_amdgcn_s_wait_asynccnt(0);
#else
    for (int idx = tid; idx < C_Hc * 18; idx += 256) {
        int ci = idx / 18, j = idx % 18;
        int tg = t0 + j - 1;
        if (tg < 0) tg = -tg;                 // reflect: -1 -> 1
        if (tg >= T) tg = 2 * T - 2 - tg;     // reflect: T -> T-2
        ldsX[idx] = xb[(long)ci * T + tg];
    }
#endif
    __syncthreads();

    int lane = tid & 31, wv = tid >> 5;
    int oc0 = blockIdx.y * 128 + wv * 16;
    int r = lane & 15, hi = lane >> 4;

    v8f acc = {0.f, 0.f, 0.f, 0.f, 0.f, 0.f, 0.f, 0.f};
    const float* Wrow = W3 + (long)(oc0 + r) * (C_Hc * 3) + hi * 2 * 3;  // [o][ci][3]
    __builtin_prefetch(Wrow, 0, 3);
    for (int k0 = 0; k0 < C_Hc; k0 += 4) {
        int kk = k0 + hi * 2;
        #pragma unroll
        for (int dt = 0; dt < 3; ++dt) {
            v2f a, bb;
            a.x = Wrow[k0 * 3 + dt];
            a.y = Wrow[k0 * 3 + 3 + dt];
            bb.x = ldsX[kk * 18 + r + dt];
            bb.y = ldsX[(kk + 1) * 18 + r + dt];
            acc = __builtin_amdgcn_wmma_f32_16x16x4_f32(false, a, false, bb, (short)0, acc,
                                                        false, false);
        }
    }

    float* yb = Y + (long)b * C_Hc * T;
    #pragma unroll
    for (int g = 0; g < 8; ++g) {
        int oc = oc0 + g + 8 * hi;
        float v = gelu_f(acc[g] + bias[oc]);
        long off = (long)oc * T + t0 + r;
        yb[off] += v;
    }
}

// ---------------- nearest-neighbor 2x upsample along T ----------------
__global__ void upsample2_kernel(const float* __restrict__ in, float* __restrict__ out, int T) {
    long gid = (long)blockIdx.x * blockDim.x + threadIdx.x;
    long total = (long)Bc * C_Hc * T;
    if (gid >= total) return;
    int t = (int)(gid % T);
    long rest = gid / T;
    float v = in[gid];
    out[rest * (2L * T) + 2 * t]     = v;
    out[rest * (2L * T) + 2 * t + 1] = v;
}

extern "C" void kernel_launch(void* const* d_in, const int* in_sizes, int n_in,
                              void* d_out, int out_size, void* d_ws, size_t ws_size,
                              hipStream_t stream) {
    (void)in_sizes; (void)n_in; (void)out_size; (void)ws_size;
    // setup_inputs() dict order: per level n: qw,kv,inw,condw ; then globals
    const float* inb     = (const float*)d_in[20];
    const float* condb   = (const float*)d_in[21];
    const float* conv3_w = (const float*)d_in[22];
    const float* conv3_b = (const float*)d_in[23];
    const float* conv1_w = (const float*)d_in[24];
    const float* conv1_b = (const float*)d_in[25];
    const float* out_w   = (const float*)d_in[26];
    const float* out_b   = (const float*)d_in[27];

    float* ws = (float*)d_ws;
    const long RS_ELEMS = (long)Bc * C_Hc * 1024;           // max token count
    float* rsA  = ws;
    float* rsB  = rsA + RS_ELEMS;
    float* Dbuf = rsB + RS_ELEMS;                           // skip / cond buffer
    float* qn   = Dbuf + RS_ELEMS;                          // [B,64,Tmax]
    float* kn   = qn + (long)Bc * C_Kc * 1024;              // [B,64,512]
    float* zb   = kn + (long)Bc * C_Kc * Sc;                // [B,cwz,T] (<=128*64)

    float* rs = rsA; float* rsOther = rsB;

    for (int n = 0; n < LVLc; ++n) {
        int cwz = 8 << (4 - n);
        int T   = 64 << n;
        const float* qw    = (const float*)d_in[4 * n + 0];
        const float* kv    = (const float*)d_in[4 * n + 1];
        const float* inw   = (const float*)d_in[4 * n + 2];
        const float* condw = (const float*)d_in[4 * n + 3];
        long qwBatch = (long)(C_Kc + cwz) * T;
        long kvBatch = (long)(C_Kc + cwz) * Sc;

        { // normalize q heads
            long total = (long)Bc * T;
            l2norm_heads_kernel<<<(int)((total + 255) / 256), 256, 0, stream>>>(qw, qn, T, qwBatch);
        }
        { // normalize k heads
            long total = (long)Bc * Sc;
            l2norm_heads_kernel<<<(int)((total + 255) / 256), 256, 0, stream>>>(kv, kn, Sc, kvBatch);
        }
        { // attention -> z
            int waves = Bc * NHEADc * T;
            attn_kernel<<<(waves + 7) / 8, 256, 0, stream>>>(qn, kn, kv, zb, T, cwz);
        }
        { // in-proj: rs (+)= inw @ w + inb[n]
            dim3 grid(Bc * (T / 16), 2);
            proj_wmma_kernel<<<grid, 256, 0, stream>>>(qw + (long)C_Kc * T, qwBatch,
                                                       inw, inb + n * C_Hc,
                                                       rs, (long)C_Hc * T,
                                                       cwz, T, C_Hc, n == 0 ? 0 : MODE_ADD);
        }
        { // cond branch: D = condw @ z + condb[n]
            dim3 grid(Bc * (T / 16), 2);
            proj_wmma_kernel<<<grid, 256, 0, stream>>>(zb, (long)cwz * T,
                                                       condw, condb + n * C_Hc,
                                                       Dbuf, (long)C_Hc * T,
                                                       cwz, T, C_Hc, 0);
        }
        { // D += gelu(conv3(reflect_pad(rs)) + b3[n])
            dim3 grid(Bc * (T / 16), 2);
            conv3_wmma_kernel<<<grid, 256, 0, stream>>>(rs, conv3_w + (long)n * C_Hc * C_Hc * 3,
                                                        conv3_b + n * C_Hc, Dbuf, T);
        }
        { // rs += gelu(conv1_w[n] @ D + conv1_b[n])
            dim3 grid(Bc * (T / 16), 2);
            proj_wmma_kernel<<<grid, 256, 0, stream>>>(Dbuf, (long)C_Hc * T,
                                                       conv1_w + (long)n * C_Hc * C_Hc,
                                                       conv1_b + n * C_Hc,
                                                       rs, (long)C_Hc * T,
                                                       C_Hc, T, C_Hc, MODE_ADD | MODE_GELU);
        }
        if (n < LVLc - 1) { // nearest 2x upsample, ping-pong buffers
            long total = (long)Bc * C_Hc * T;
            upsample2_kernel<<<(int)((total + 255) / 256), 256, 0, stream>>>(rs, rsOther, T);
            float* tmp = rs; rs = rsOther; rsOther = tmp;
        }
    }

    { // final: out = out_w @ rs + out_b, OC = 80
        int T = 1024;
        dim3 grid(Bc * (T / 16), 1);
        proj_wmma_kernel<<<grid, 256, 0, stream>>>(rs, (long)C_Hc * T, out_w, out_b,
                                                   (float*)d_out, (long)C_OUTc * T,
                                                   C_Hc, T, C_OUTc, 0);
    }
}